// SAGE_51092930953818
// MI455X (gfx1250) — compile-verified
//
#include <hip/hip_runtime.h>
#include <hip/hip_bf16.h>

#define N_NODES 100000
#define N_EDGES 1280000
#define D 64
#define LDP 68   // padded LDS row stride (floats): bank step = 4 -> conflict-free

typedef float v2f __attribute__((ext_vector_type(2)));
typedef float v8f __attribute__((ext_vector_type(8)));

// ---------------------------------------------------------------------------
// Zero a float buffer (one float4 per thread, scalar tail)
// ---------------------------------------------------------------------------
__global__ void sage_zero_kernel(float* __restrict__ p, int n) {
    int i = (blockIdx.x * blockDim.x + threadIdx.x) * 4;
    if (i + 3 < n) {
        *(float4*)(p + i) = make_float4(0.f, 0.f, 0.f, 0.f);
    } else {
        for (int j = i; j < n; ++j) p[j] = 0.f;
    }
}

// ---------------------------------------------------------------------------
// Edge scatter: one wave (32 lanes) per edge. Each lane adds a float2 slice
// of x[src] into agg[dst] with global_atomic_add_f32 (coalesced 256B/wave).
// Lane 0 optionally accumulates the in-degree count.
// ---------------------------------------------------------------------------
__global__ __launch_bounds__(256) void sage_scatter_kernel(
    const float* __restrict__ feat, const int* __restrict__ src,
    const int* __restrict__ dst, float* __restrict__ agg,
    float* __restrict__ cnt, int n_edges, int do_count)
{
    int gid  = blockIdx.x * blockDim.x + threadIdx.x;
    int edge = gid >> 5;
    int lane = gid & 31;
    if (edge >= n_edges) return;
    int s = src[edge];
    int d = dst[edge];
    const float2 v = *(const float2*)(feat + (size_t)s * D + lane * 2);
    float* ap = agg + (size_t)d * D + lane * 2;
    atomicAdd(ap, v.x);
    atomicAdd(ap + 1, v.y);
    if (do_count && lane == 0) atomicAdd(cnt + d, 1.0f);
}

// ---------------------------------------------------------------------------
// Fused SAGE layer:  out = mean(agg,cnt) @ Wl^T + b + xin @ Wr^T  [+ReLU]
// Block = 128 threads = 4 waves; each wave owns a 16-node tile and computes
// its full 16x64 output with f32 WMMA (16x16x4), accumulating both matmuls
// into the same C fragments. Weights + per-wave mean tile staged in LDS.
// ---------------------------------------------------------------------------
__global__ __launch_bounds__(128) void sage_layer_wmma_kernel(
    const float* __restrict__ xin,   // [N,64] this layer's input features
    const float* __restrict__ agg,   // [N,64] neighbor feature sums
    const float* __restrict__ cnt,   // [N]    in-degree counts
    const float* __restrict__ Wl,    // [64,64] row-major (out,in)
    const float* __restrict__ bias,  // [64]
    const float* __restrict__ Wr,    // [64,64] row-major (out,in)
    float* __restrict__ out,         // [N,64]
    int n_nodes, int do_relu)
{
    __shared__ float s_wl[D * LDP];          // Wl, padded rows
    __shared__ float s_wr[D * LDP];          // Wr, padded rows
    __shared__ float s_b[D];
    __shared__ float s_mean[4][16 * LDP];    // per-wave mean tile

    const int tid  = threadIdx.x;
    const int wave = tid >> 5;
    const int lane = tid & 31;

    // Stage both weight matrices (+bias) into LDS, padded to LDP.
    for (int i = tid; i < D * D; i += 128) {
        int n = i >> 6, k = i & 63;
        s_wl[n * LDP + k] = Wl[i];
        s_wr[n * LDP + k] = Wr[i];
    }
    if (tid < D) s_b[tid] = bias[tid];
    __syncthreads();

    const int n_tiles = (n_nodes + 15) >> 4;
    const int tile    = blockIdx.x * 4 + wave;
    if (tile >= n_tiles) return;             // whole-wave uniform
    const int row0 = tile << 4;

    // Stage this wave's 16x64 mean tile: mean = agg / max(cnt,1)
    float* sm = &s_mean[wave][0];
    #pragma unroll 4
    for (int r = 0; r < 16; ++r) {
        int row = row0 + r;                  // uniform across lanes
        float2 v = make_float2(0.f, 0.f);
        float  c = 1.0f;
        if (row < n_nodes) {
            c = cnt[row];
            if (c < 1.0f) c = 1.0f;
            v = *(const float2*)(agg + (size_t)row * D + lane * 2);
        }
        float inv = 1.0f / c;
        sm[r * LDP + lane * 2]     = v.x * inv;
        sm[r * LDP + lane * 2 + 1] = v.y * inv;
    }
    // Same-wave LDS cross-lane visibility before fragment reads.
    asm volatile("s_wait_dscnt 0" ::: "memory");

    // WMMA fragment indexing (ISA 7.12.2, 32-bit A 16x4 / C 16x16):
    //   lane m = lane&15, half-wave selects K pair (khalf = (lane>>4)*2)
    const int m     = lane & 15;
    const int khalf = (lane >> 4) << 1;
    const int xrow  = (row0 + m < n_nodes) ? (row0 + m) : (n_nodes - 1);

    v8f acc[4];
    #pragma unroll
    for (int t = 0; t < 4; ++t) acc[t] = (v8f){0.f,0.f,0.f,0.f,0.f,0.f,0.f,0.f};

    #pragma unroll 4
    for (int kk = 0; kk < D; kk += 4) {
        const int k0 = kk + khalf;
        // A fragments: mean (from LDS) and x (from global, rows L2-hot)
        v2f a_mean = { sm[m * LDP + k0], sm[m * LDP + k0 + 1] };
        const float* xr = xin + (size_t)xrow * D + k0;
        v2f a_x = { xr[0], xr[1] };
        #pragma unroll
        for (int t = 0; t < 4; ++t) {
            const int n = t * 16 + m;        // output column this lane supplies
            // B = W^T : frag = { W[n][k0], W[n][k0+1] }
            v2f bl = { s_wl[n * LDP + k0], s_wl[n * LDP + k0 + 1] };
            v2f br = { s_wr[n * LDP + k0], s_wr[n * LDP + k0 + 1] };
            acc[t] = __builtin_amdgcn_wmma_f32_16x16x4_f32(
                false, a_mean, false, bl, (short)0, acc[t], false, false);
            acc[t] = __builtin_amdgcn_wmma_f32_16x16x4_f32(
                false, a_x,    false, br, (short)0, acc[t], false, false);
        }
    }

    // Epilogue: + bias, optional ReLU, store.
    // C/D layout: VGPR v holds row (v + 8*(lane>>4)), col = lane&15 (per tile).
    const int rbase = row0 + ((lane >> 4) << 3);
    #pragma unroll
    for (int t = 0; t < 4; ++t) {
        const int n  = t * 16 + m;
        const float bv = s_b[n];
        #pragma unroll
        for (int v = 0; v < 8; ++v) {
            float val = acc[t][v] + bv;
            if (do_relu) val = val > 0.f ? val : 0.f;
            int row = rbase + v;
            if (row < n_nodes) out[(size_t)row * D + n] = val;
        }
    }
}

// ---------------------------------------------------------------------------
// Launch: zero -> scatter -> layer1(ReLU) -> zero -> scatter -> layer2
// ---------------------------------------------------------------------------
extern "C" void kernel_launch(void* const* d_in, const int* in_sizes, int n_in,
                              void* d_out, int out_size, void* d_ws, size_t ws_size,
                              hipStream_t stream) {
    const float* x   = (const float*)d_in[0];
    const int*   ei  = (const int*)d_in[1];        // [2, E] edge index
    const float* W1l = (const float*)d_in[2];
    const float* b1  = (const float*)d_in[3];
    const float* W1r = (const float*)d_in[4];
    const float* W2l = (const float*)d_in[5];
    const float* b2  = (const float*)d_in[6];
    const float* W2r = (const float*)d_in[7];
    float* out = (float*)d_out;

    const int* src = ei;
    const int* dst = ei + N_EDGES;

    float* agg = (float*)d_ws;                     // N*64 floats
    float* cnt = agg + (size_t)N_NODES * D;        // N floats
    float* h   = cnt + N_NODES;                    // N*64 floats

    const int zero_n1 = N_NODES * D + N_NODES;     // agg + cnt together
    const int zero_n2 = N_NODES * D;               // agg only
    dim3 zb(256);
    dim3 zg1((zero_n1 / 4 + 255) / 256), zg2((zero_n2 / 4 + 255) / 256);

    dim3 sb(256), sg((N_EDGES * 32 + 255) / 256);  // wave per edge
    dim3 lb(128);
    dim3 lg((((N_NODES + 15) / 16) + 3) / 4);      // 4 node-tiles per block

    // ---- layer 1 ----
    sage_zero_kernel<<<zg1, zb, 0, stream>>>(agg, zero_n1);
    sage_scatter_kernel<<<sg, sb, 0, stream>>>(x, src, dst, agg, cnt, N_EDGES, 1);
    sage_layer_wmma_kernel<<<lg, lb, 0, stream>>>(x, agg, cnt, W1l, b1, W1r,
                                                  h, N_NODES, 1);
    // ---- layer 2 (reuse degree counts) ----
    sage_zero_kernel<<<zg2, zb, 0, stream>>>(agg, zero_n2);
    sage_scatter_kernel<<<sg, sb, 0, stream>>>(h, src, dst, agg, cnt, N_EDGES, 0);
    sage_layer_wmma_kernel<<<lg, lb, 0, stream>>>(h, agg, cnt, W2l, b2, W2r,
                                                  out, N_NODES, 0);
}